// NASOP_45792941310621
// MI455X (gfx1250) — compile-verified
//
#include <hip/hip_runtime.h>

typedef float v2f __attribute__((ext_vector_type(2)));
typedef float v8f __attribute__((ext_vector_type(8)));

#define FEAT 128

// ---------------------------------------------------------------------------
// Kernel 1: h = x @ W^T + b  via V_WMMA_F32_16X16X4_F32, fp32 end-to-end.
// Also initializes out = h (this is exactly the self-loop contribution,
// since reference appends identity edges and scatter-adds h[i] into out[i]).
//
// Tile: one wave -> one 16(M) x 16(N) tile; 8 waves/block cover all 128 cols
// for a 16-row slab. K loop: 128/4 = 32 wmma ops per tile.
//
// ISA VGPR layouts (CDNA5 ISA):
//   A 16x4 f32 : lane<16 holds M=lane, VGPR0=K0,VGPR1=K1; lane>=16 K2/K3.
//   B 4x16 f32 : mirrored, N = lane&15.   B[k][n] = W^T[k][n] = W[n][k],
//                so each lane reads a contiguous float2 from row (col0+n) of W.
//   C/D 16x16  : VGPR r -> M = r + 8*(lane>=16), N = lane&15.
// ---------------------------------------------------------------------------
__global__ __launch_bounds__(256)
void gcn_gemm_wmma(const float* __restrict__ x, const float* __restrict__ W,
                   const float* __restrict__ bias, float* __restrict__ h,
                   float* __restrict__ out, int n_nodes) {
  const int lane = threadIdx.x & 31;
  const int wave = threadIdx.x >> 5;          // 0..7 -> which 16-col tile
  const int row0 = blockIdx.x * 16;
  if (row0 >= n_nodes) return;                // uniform per block

  const int col0 = wave * 16;
  const int half = lane >> 4;                 // 0: K0/K1 pair, 1: K2/K3 pair
  const int l16  = lane & 15;                 // A row / B col owned by lane
  const int koff = half * 2;

  const float* __restrict__ arow = x + (size_t)(row0 + l16) * FEAT + koff;
  const float* __restrict__ brow = W + (size_t)(col0 + l16) * FEAT + koff;

  // bias depends only on N=l16: bake it into the accumulator init
  const float bv = bias[col0 + l16];
  v8f c = {bv, bv, bv, bv, bv, bv, bv, bv};

#pragma unroll
  for (int k0 = 0; k0 < FEAT; k0 += 4) {
    v2f a = *(const v2f*)(arow + k0);   // A[M=l16][k0+koff .. +1]
    v2f b = *(const v2f*)(brow + k0);   // B[k0+koff .. +1][N=l16]
    c = __builtin_amdgcn_wmma_f32_16x16x4_f32(
        /*neg_a=*/false, a, /*neg_b=*/false, b,
        /*c_mod=*/(short)0, c, /*reuse_a=*/false, /*reuse_b=*/false);
  }

  // Store D: element r lives at M = half*8 + r, N = l16.
  const size_t base = (size_t)(row0 + half * 8) * FEAT + col0 + l16;
#pragma unroll
  for (int r = 0; r < 8; ++r) {
    const float v = c[r];
    h[base + (size_t)r * FEAT]   = v;   // transformed features (gather source)
    out[base + (size_t)r * FEAT] = v;   // self-loop contribution pre-seeded
  }
}

// ---------------------------------------------------------------------------
// Kernel 2: edge scatter-add.  One wave per edge: lane L gathers float4
// h[src][4L..4L+3] and atomically accumulates into out[dst].  h and out are
// L2-resident (25.6 MB each << 192 MB L2); the int64 edge stream is read
// non-temporally so the one-shot 12.8 MB stream doesn't evict them.
// ---------------------------------------------------------------------------
__global__ __launch_bounds__(256)
void gcn_scatter(const long long* __restrict__ ei, const float* __restrict__ h,
                 float* __restrict__ out, int n_edges) {
  const int lane = threadIdx.x & 31;
  const long long e = (long long)blockIdx.x * 8 + (threadIdx.x >> 5);
  if (e >= n_edges) return;                   // uniform per wave

  const int src = (int)__builtin_nontemporal_load(ei + e);
  const int dst = (int)__builtin_nontemporal_load(ei + (long long)n_edges + e);

  const float4 v = *(const float4*)(h + (size_t)src * FEAT + lane * 4);
  float* o = out + (size_t)dst * FEAT + lane * 4;
  atomicAdd(o + 0, v.x);
  atomicAdd(o + 1, v.y);
  atomicAdd(o + 2, v.z);
  atomicAdd(o + 3, v.w);
}

// ---------------------------------------------------------------------------
// Launch: GEMM (+ self-loop init of out) then edge scatter; stream order
// provides the dependency.  h lives in d_ws (needs 50000*128*4 = 25.6 MB).
// ---------------------------------------------------------------------------
extern "C" void kernel_launch(void* const* d_in, const int* in_sizes, int n_in,
                              void* d_out, int out_size, void* d_ws, size_t ws_size,
                              hipStream_t stream) {
  const float*      x  = (const float*)d_in[0];      // [N,128] f32
  const long long*  ei = (const long long*)d_in[1];  // [2,E]  int64
  const float*      W  = (const float*)d_in[2];      // [128,128] f32
  const float*      b  = (const float*)d_in[3];      // [128] f32
  float* out = (float*)d_out;                        // [N,128] f32
  float* h   = (float*)d_ws;                         // [N,128] f32 scratch

  const int n_nodes = in_sizes[0] / FEAT;            // 50000
  const int n_edges = in_sizes[1] / 2;               // 800000

  dim3 gemm_grid((n_nodes + 15) / 16);
  gcn_gemm_wmma<<<gemm_grid, 256, 0, stream>>>(x, W, b, h, out, n_nodes);

  dim3 scat_grid((unsigned)((n_edges + 7) / 8));     // 8 waves (edges) / block
  gcn_scatter<<<scat_grid, 256, 0, stream>>>(ei, h, out, n_edges);
}